// Model_36481452212490
// MI455X (gfx1250) — compile-verified
//
#include <hip/hip_runtime.h>
#include <hip/hip_bf16.h>

typedef __attribute__((ext_vector_type(2))) float v2f;
typedef __attribute__((ext_vector_type(8))) float v8f;
typedef int v4i __attribute__((vector_size(16)));

#if defined(__gfx1250__) && __has_builtin(__builtin_amdgcn_global_load_async_to_lds_b128)
#define USE_ASYNC_LDS 1
#else
#define USE_ASYNC_LDS 0
#endif

#if USE_ASYNC_LDS
__device__ __forceinline__ void async_cp16(const float* g, float* l) {
    __builtin_amdgcn_global_load_async_to_lds_b128(
        (__attribute__((address_space(1))) v4i*)g,
        (__attribute__((address_space(3))) v4i*)l, 0, 0);
}
__device__ __forceinline__ void wait_async0() {
#if __has_builtin(__builtin_amdgcn_s_wait_asynccnt)
    __builtin_amdgcn_s_wait_asynccnt(0);
#else
    asm volatile("s_wait_asynccnt 0x0" ::: "memory");
#endif
}
#endif

// ---------------------------------------------------------------------------
// K1: dis[b,i] = rsqrt(max(1, 1 + sum_j A[b,i,j]))   (wave-per-row reduction)
// ---------------------------------------------------------------------------
__global__ void deg_dis_kernel(const float* __restrict__ A, float* __restrict__ dis) {
    int gwid = (blockIdx.x * blockDim.x + threadIdx.x) >> 5;  // row id in [0, B*1024)
    int lane = threadIdx.x & 31;
    const float4* row = (const float4*)(A + ((size_t)gwid << 10));
    float s = 0.f;
#pragma unroll
    for (int it = 0; it < 8; ++it) {
        float4 v = row[it * 32 + lane];
        s += v.x + v.y + v.z + v.w;
    }
#pragma unroll
    for (int off = 16; off; off >>= 1) s += __shfl_xor(s, off, 32);
    if (lane == 0) dis[gwid] = rsqrtf(fmaxf(1.0f + s, 1.0f));
}

// ---------------------------------------------------------------------------
// K2: fused conv(3/5/7) -> +bias -> ReLU -> GEMM with Wg1 [1008,40]
//     output H1s[b,j,h] = dis[b,j] * (x_conv @ Wg1)[b,j,h], stride 48, cols
//     40..47 zeroed (padding for WMMA B-matrix).
// ---------------------------------------------------------------------------
#define CH 84  // Wg1 d-chunk (1008 = 12*84)
__global__ void conv_gemm1_kernel(const float* __restrict__ x,
                                  const float* __restrict__ W1, const float* __restrict__ b1,
                                  const float* __restrict__ W2, const float* __restrict__ b2,
                                  const float* __restrict__ W3, const float* __restrict__ b3,
                                  const float* __restrict__ Wg1,
                                  const float* __restrict__ dis,
                                  float* __restrict__ H1s) {
    __shared__ float xl[32 * 256];   // x window per node
    __shared__ float wg[CH * 40];    // Wg1 chunk
    __shared__ float cw[216];        // conv weights+biases

    int tid = threadIdx.x;
    int blk = blockIdx.x;            // 64 blocks
    int b = blk >> 2;
    int n = ((blk & 3) << 8) + tid;  // node

#pragma unroll
    for (int w = 0; w < 32; ++w)
        xl[w * 256 + tid] = x[((size_t)(b * 32 + w) << 10) + n];

    if (tid < 36) cw[tid] = W1[tid];
    if (tid < 60) cw[36 + tid] = W2[tid];
    if (tid < 84) cw[96 + tid] = W3[tid];
    if (tid < 12) { cw[180 + tid] = b1[tid]; cw[192 + tid] = b2[tid]; cw[204 + tid] = b3[tid]; }

    float4 acc[10];
#pragma unroll
    for (int q = 0; q < 10; ++q) acc[q] = make_float4(0.f, 0.f, 0.f, 0.f);

    for (int ch = 0; ch < 12; ++ch) {
        int d0 = ch * CH;
        __syncthreads();
        for (int t = tid; t < CH * 10; t += 256)   // 840 float4
            ((float4*)wg)[t] = ((const float4*)(Wg1 + (size_t)d0 * 40))[t];
        __syncthreads();
        for (int dd = 0; dd < CH; ++dd) {
            int d = d0 + dd;
            float v;
            if (d < 360) {                       // k=3, L=30
                int c = d / 30, t = d - c * 30;
                const float* w = cw + c * 3;
                v = cw[180 + c] + w[0] * xl[t * 256 + tid] + w[1] * xl[(t + 1) * 256 + tid]
                    + w[2] * xl[(t + 2) * 256 + tid];
            } else if (d < 696) {                // k=5, L=28
                int d2 = d - 360;
                int c = d2 / 28, t = d2 - c * 28;
                const float* w = cw + 36 + c * 5;
                v = cw[192 + c];
#pragma unroll
                for (int u = 0; u < 5; ++u) v += w[u] * xl[(t + u) * 256 + tid];
            } else {                             // k=7, L=26
                int d2 = d - 696;
                int c = d2 / 26, t = d2 - c * 26;
                const float* w = cw + 96 + c * 7;
                v = cw[204 + c];
#pragma unroll
                for (int u = 0; u < 7; ++u) v += w[u] * xl[(t + u) * 256 + tid];
            }
            v = fmaxf(v, 0.f);
            const float4* wrow = (const float4*)(wg + dd * 40);
#pragma unroll
            for (int q = 0; q < 10; ++q) {
                float4 t4 = wrow[q];
                acc[q].x += v * t4.x; acc[q].y += v * t4.y;
                acc[q].z += v * t4.z; acc[q].w += v * t4.w;
            }
        }
    }

    float dv = dis[(b << 10) + n];
    float4* o = (float4*)(H1s + ((size_t)(b << 10) + n) * 48);
#pragma unroll
    for (int q = 0; q < 10; ++q)
        o[q] = make_float4(dv * acc[q].x, dv * acc[q].y, dv * acc[q].z, dv * acc[q].w);
    o[10] = make_float4(0.f, 0.f, 0.f, 0.f);
    o[11] = make_float4(0.f, 0.f, 0.f, 0.f);
}

// ---------------------------------------------------------------------------
// K3/K5: out[b,i,h<nvalid] = act( dis[i]*( sum_k A[b,i,k]*Hs[b,k,h] + Hs[b,i,h] ) + bias[h] )
//        WMMA f32 16x16x4, double-buffered LDS tiles filled with
//        GLOBAL_LOAD_ASYNC_TO_LDS_B128 (ASYNCcnt) when available.
//        Block = 256 thr (8 waves), 128 rows x NT*16 cols.
// ---------------------------------------------------------------------------
template <int NT>
__global__ void prop_kernel(const float* __restrict__ A, const float* __restrict__ Hs,
                            const float* __restrict__ dis, const float* __restrict__ bias,
                            int nvalid, int doRelu, float* __restrict__ out) {
    constexpr int BS = NT * 16;
    __shared__ float Al[2][128 * 36];   // 32-col K panel, padded stride 36
    __shared__ float Bl[2][32 * BS];

    int tid = threadIdx.x;
    int w = tid >> 5, lane = tid & 31;
    int b = blockIdx.y;
    int Mbase = blockIdx.x * 128;
    const float* Ab = A + ((size_t)b << 20);
    const float* Hb = Hs + ((size_t)(b << 10)) * BS;

    v8f cacc[NT];
#pragma unroll
    for (int nt = 0; nt < NT; ++nt)
#pragma unroll
        for (int e = 0; e < 8; ++e) cacc[nt][e] = 0.f;

    auto prefetch = [&](int k0, float* Ad, float* Bd) {
#pragma unroll
        for (int it = 0; it < 4; ++it) {           // A tile: 128 x 32 (16 KB)
            int idx = it * 256 + tid;
            int r = idx >> 3, c4 = (idx & 7) << 2;
            const float* src = Ab + ((size_t)(Mbase + r) << 10) + k0 + c4;
#if USE_ASYNC_LDS
            async_cp16(src, Ad + r * 36 + c4);
#else
            *(float4*)(Ad + r * 36 + c4) = *(const float4*)src;
#endif
        }
        for (int t = tid; t < 8 * BS; t += 256) {  // B tile: 32 x BS
            int r = t / (BS / 4), c4 = (t % (BS / 4)) << 2;
            const float* src = Hb + (size_t)(k0 + r) * BS + c4;
#if USE_ASYNC_LDS
            async_cp16(src, Bd + r * BS + c4);
#else
            *(float4*)(Bd + r * BS + c4) = *(const float4*)src;
#endif
        }
    };

    prefetch(0, Al[0], Bl[0]);

    for (int i = 0; i < 32; ++i) {       // 32 K-steps of 32
        int cur = i & 1;
#if USE_ASYNC_LDS
        wait_async0();                   // my async tile-writes landed in LDS
#endif
        __syncthreads();                 // everyone's writes landed; prev compute done
        if (i + 1 < 32) prefetch((i + 1) * 32, Al[cur ^ 1], Bl[cur ^ 1]);

        const float* Ac = Al[cur];
        const float* Bc = Bl[cur];
#pragma unroll
        for (int kk = 0; kk < 32; kk += 4) {
            // A frag (16x4 f32): lanes 0-15 -> K={kk,kk+1}, lanes 16-31 -> K={kk+2,kk+3}
            int ar = w * 16 + (lane & 15);
            int ac = kk + ((lane >> 4) << 1);
            v2f a;
            a.x = Ac[ar * 36 + ac];
            a.y = Ac[ar * 36 + ac + 1];
#pragma unroll
            for (int nt = 0; nt < NT; ++nt) {
                int nn = nt * 16 + (lane & 15);
                v2f bf;
                bf.x = Bc[ac * BS + nn];
                bf.y = Bc[(ac + 1) * BS + nn];
                cacc[nt] = __builtin_amdgcn_wmma_f32_16x16x4_f32(
                    false, a, false, bf, (short)0, cacc[nt], false, false);
            }
        }
    }

    // epilogue: +identity term, *dis[i], +bias, optional ReLU
    int half = lane >> 4;
#pragma unroll
    for (int nt = 0; nt < NT; ++nt) {
        int h = nt * 16 + (lane & 15);
        float bi = (h < nvalid) ? bias[h] : 0.f;
#pragma unroll
        for (int r = 0; r < 8; ++r) {
            int i = Mbase + w * 16 + r + half * 8;
            float val = cacc[nt][r] + Hb[(size_t)i * BS + h];
            val = dis[(b << 10) + i] * val + bi;
            if (doRelu) val = fmaxf(val, 0.f);
            if (h < nvalid) out[((size_t)(b << 10) + i) * BS + h] = val;
        }
    }
}

// ---------------------------------------------------------------------------
// K4: H2s[j, h2<10] = dis[j] * (X1[j,0:40] @ Wg2[40,10]); cols 10..15 zeroed
// ---------------------------------------------------------------------------
__global__ void gemm2_kernel(const float* __restrict__ X1, const float* __restrict__ Wg2,
                             const float* __restrict__ dis, float* __restrict__ H2s) {
    __shared__ float w2[400];
    int tid = threadIdx.x;
    for (int t = tid; t < 400; t += 256) w2[t] = Wg2[t];
    __syncthreads();
    int j = blockIdx.x * 256 + tid;
    const float4* xr4 = (const float4*)(X1 + (size_t)j * 48);
    float xr[40];
#pragma unroll
    for (int q = 0; q < 10; ++q) {
        float4 v = xr4[q];
        xr[q * 4 + 0] = v.x; xr[q * 4 + 1] = v.y; xr[q * 4 + 2] = v.z; xr[q * 4 + 3] = v.w;
    }
    float dv = dis[j];
    float* o = H2s + (size_t)j * 16;
#pragma unroll
    for (int h2 = 0; h2 < 10; ++h2) {
        float s = 0.f;
#pragma unroll
        for (int h = 0; h < 40; ++h) s += xr[h] * w2[h * 10 + h2];
        o[h2] = dv * s;
    }
#pragma unroll
    for (int h2 = 10; h2 < 16; ++h2) o[h2] = 0.f;
}

// ---------------------------------------------------------------------------
// K6a: H3s[j] = dis[j] * (X2[j,0:10] @ Wg3[10,1])
// ---------------------------------------------------------------------------
__global__ void gemm3_kernel(const float* __restrict__ X2, const float* __restrict__ Wg3,
                             const float* __restrict__ dis, float* __restrict__ H3s) {
    int j = blockIdx.x * 256 + threadIdx.x;
    const float* xr = X2 + (size_t)j * 16;
    float s = 0.f;
#pragma unroll
    for (int h = 0; h < 10; ++h) s += xr[h] * Wg3[h];
    H3s[j] = dis[j] * s;
}

// ---------------------------------------------------------------------------
// K6b: out[b,i] = dis[i]*( sum_k A[b,i,k]*H3s[b,k] + H3s[b,i] ) + bg3
// ---------------------------------------------------------------------------
__global__ void prop3_kernel(const float* __restrict__ A, const float* __restrict__ H3s,
                             const float* __restrict__ dis, const float* __restrict__ bg3,
                             float* __restrict__ out) {
    int gwid = (blockIdx.x * blockDim.x + threadIdx.x) >> 5;
    int lane = threadIdx.x & 31;
    int b = gwid >> 10, i = gwid & 1023;
    const float4* row = (const float4*)(A + ((size_t)gwid << 10));
    const float* h = H3s + (b << 10);
    const float4* h4 = (const float4*)h;
    float s = 0.f;
#pragma unroll
    for (int it = 0; it < 8; ++it) {
        int k4 = it * 32 + lane;
        float4 av = row[k4];
        float4 hv = h4[k4];
        s += av.x * hv.x + av.y * hv.y + av.z * hv.z + av.w * hv.w;
    }
#pragma unroll
    for (int off = 16; off; off >>= 1) s += __shfl_xor(s, off, 32);
    if (lane == 0) out[gwid] = dis[gwid] * (s + h[i]) + bg3[0];
}

// ---------------------------------------------------------------------------
extern "C" void kernel_launch(void* const* d_in, const int* in_sizes, int n_in,
                              void* d_out, int out_size, void* d_ws, size_t ws_size,
                              hipStream_t stream) {
    const float* x   = (const float*)d_in[0];
    const float* A   = (const float*)d_in[1];
    const float* W1  = (const float*)d_in[2];
    const float* b1  = (const float*)d_in[3];
    const float* W2  = (const float*)d_in[4];
    const float* b2  = (const float*)d_in[5];
    const float* W3  = (const float*)d_in[6];
    const float* b3  = (const float*)d_in[7];
    const float* Wg1 = (const float*)d_in[8];
    const float* bg1 = (const float*)d_in[9];
    const float* Wg2 = (const float*)d_in[10];
    const float* bg2 = (const float*)d_in[11];
    const float* Wg3 = (const float*)d_in[12];
    const float* bg3 = (const float*)d_in[13];
    float* out = (float*)d_out;

    float* ws  = (float*)d_ws;
    float* dis = ws;                   // 16384
    float* H1s = dis + 16384;          // 16*1024*48
    float* X1  = H1s + 16 * 1024 * 48; // 16*1024*48
    float* H2s = X1 + 16 * 1024 * 48;  // 16*1024*16
    float* X2  = H2s + 16 * 1024 * 16; // 16*1024*16
    float* H3s = X2 + 16 * 1024 * 16;  // 16384

    deg_dis_kernel<<<2048, 256, 0, stream>>>(A, dis);
    conv_gemm1_kernel<<<64, 256, 0, stream>>>(x, W1, b1, W2, b2, W3, b3, Wg1, dis, H1s);
    prop_kernel<3><<<dim3(8, 16), 256, 0, stream>>>(A, H1s, dis, bg1, 40, 1, X1);
    gemm2_kernel<<<64, 256, 0, stream>>>(X1, Wg2, dis, H2s);
    prop_kernel<1><<<dim3(8, 16), 256, 0, stream>>>(A, H2s, dis, bg2, 10, 1, X2);
    gemm3_kernel<<<64, 256, 0, stream>>>(X2, Wg3, dis, H3s);
    prop3_kernel<<<2048, 256, 0, stream>>>(A, H3s, dis, bg3, out);
}